// MergeModel_4638564680428
// MI455X (gfx1250) — compile-verified
//
#include <hip/hip_runtime.h>
#include <hip/hip_bf16.h>
#include <math.h>

// Problem dims (fixed by reference)
#define B_      8
#define E_      64
#define N_      64
#define DIN_    768
#define F_      256
#define HEADS_  8
#define DH_     32
#define LAYERS_ 2
#define HID_    256

typedef __attribute__((ext_vector_type(2))) float v2f;
typedef __attribute__((ext_vector_type(8))) float v8f;
typedef unsigned int v4u __attribute__((ext_vector_type(4)));
typedef int v8i __attribute__((ext_vector_type(8)));
typedef int v4i __attribute__((ext_vector_type(4)));

// Native fp32 WMMA: D(16x16,f32) = A(16x4,f32) * B(4x16,f32) + C
__device__ __forceinline__ v8f wmma4(v2f a, v2f b, v8f c) {
  return __builtin_amdgcn_wmma_f32_16x16x4_f32(false, a, false, b, (short)0, c,
                                               false, false);
}

// ---------------------------------------------------------------------------
// Tensor Data Mover: 2D tile (nrows x rowlen f32) global -> LDS, with
// hardware pad-on-load of 4 DWORDs after every 256 DWORDs stored
// (pad_interval=256 -> enc 7, pad_amount=4 -> enc 3). For rowlen a multiple
// of 256 this yields LDS row stride = rowlen + 4*(rowlen/256), i.e. the
// bank-conflict-free layouts used below. D# per CDNA5 ISA ch.8.
// Call from ONE wave only (TDM ignores EXEC; issues once per wave).
// This toolchain exposes the 6-arg builtin:
//   (uint32x4 g0, int32x8 g1, int32x4 g2, int32x4 g3, int32x8 extra, i32 cpol)
// ---------------------------------------------------------------------------
__device__ __forceinline__ void tdm_load_2d_pad4(const void* gsrc, void* lds_dst,
                                                 unsigned rowlen, unsigned nrows,
                                                 unsigned stride_elems) {
  const unsigned long long ga = (unsigned long long)(uintptr_t)gsrc;
  const unsigned lds = (unsigned)(uintptr_t)lds_dst;  // low 32b = LDS offset
  v4u g0;
  g0.x = 1u;                                           // count=1 (valid D#)
  g0.y = lds;                                          // lds_addr
  g0.z = (unsigned)ga;                                 // global_addr[31:0]
  g0.w = (unsigned)((ga >> 32) & 0x1FFFFFFu)           // global_addr[56:32]
         | (2u << 30);                                 // type=2 ("image")
  v8i g1;
  g1[0] = (int)((2u << 16)        // data_size = 4 bytes
                | (1u << 20)      // pad_enable
                | (7u << 22)      // pad_interval: 256 DWORDs
                | (3u << 25));    // pad_amount: 4 DWORDs
  g1[1] = (int)((rowlen & 0xFFFFu) << 16);             // abar=0 | tensor_dim0.lo
  g1[2] = (int)(((rowlen >> 16) & 0xFFFFu)             // tensor_dim0.hi
                | ((nrows & 0xFFFFu) << 16));          // tensor_dim1.lo
  g1[3] = (int)(((nrows >> 16) & 0xFFFFu)              // tensor_dim1.hi
                | ((rowlen & 0xFFFFu) << 16));         // tile_dim0
  g1[4] = (int)(nrows & 0xFFFFu);                      // tile_dim1 | tile_dim2=0
  g1[5] = (int)stride_elems;                           // tensor_dim0_stride.lo32
  g1[6] = 0;                                           // stride0.hi | stride1.lo
  g1[7] = 0;
  const v4i z4 = {0, 0, 0, 0};                         // 2D: groups 2/3 unused
  const v8i z8 = {0, 0, 0, 0, 0, 0, 0, 0};
  __builtin_amdgcn_tensor_load_to_lds(g0, g1, z4, z4, z8, 0);
}

// ---------------------------------------------------------------------------
// Kernel 1: FC  —  C[32768x256] = A[32768x768] * W[768x256] + bias
// Block: 256 threads (8 waves). One 16-row tile per block; wave w handles
// col tile blockIdx.y*8 + w. A tile DMA'd into LDS by the TDM with pad-4
// layout: LDS row stride 780 (780%64==12 -> conflict-free), element (r,k)
// at r*780 + k + 4*(k>>8).
// ---------------------------------------------------------------------------
#define LDA_ 780
__global__ __launch_bounds__(256) void fc_kernel(const float* __restrict__ A,
                                                 const float* __restrict__ W,
                                                 const float* __restrict__ bias,
                                                 float* __restrict__ C) {
  __shared__ float lA[16 * LDA_];
  const int tid  = threadIdx.x;
  const int lane = tid & 31;
  const int wave = tid >> 5;
  const int m0   = blockIdx.x * 16;
  const int n0   = (blockIdx.y * 8 + wave) * 16;

  if (tid < 32) {  // wave 0 drives the DMA
    tdm_load_2d_pad4(A + (size_t)m0 * DIN_, lA, DIN_, 16, DIN_);
    __builtin_amdgcn_s_wait_tensorcnt(0);
  }
  __syncthreads();

  const int mA = lane & 15;          // A row within tile
  const int kk = (lane >> 4) * 2;    // K sub-offset (lanes 16-31 -> K2/K3)
  const int nB = lane & 15;          // B/C column within tile

  v8f acc = {};
  const float* arow = lA + mA * LDA_ + kk;
  const float* wcol = W + n0 + nB;
  for (int k = 0; k < DIN_; k += 4) {
    const int ko = k + ((k >> 8) << 2);  // skip TDM pad DWORDs
    v2f a, b;
    a.x = arow[ko];
    a.y = arow[ko + 1];
    b.x = wcol[(size_t)(k + kk) * F_];
    b.y = wcol[(size_t)(k + kk + 1) * F_];
    acc = wmma4(a, b, acc);
  }
  const float bv  = bias[n0 + nB];
  const int mbase = m0 + (lane >> 4) * 8;
#pragma unroll
  for (int r = 0; r < 8; ++r)
    C[(size_t)(mbase + r) * F_ + n0 + nB] = acc[r] + bv;
}

// ---------------------------------------------------------------------------
// Kernel 2: two-layer GAT over one 64-node graph per workgroup (in-place).
// 256 threads = 8 waves; wave == head. x and h in LDS; x tile DMA'd in by
// the TDM (pad-4 layout reproduces LDX_=260 exactly: 256 DW/row + 4 pad).
// Phase 1: h = x*W[l,head]          (WMMA f32 16x16x4)
// Phase 2: es/ed = h . a_src/a_dst
// Phase 3: masked-softmax row stats
// Phase 4: out = softmax(attn)*h via WMMA, A fragment built on the fly; ELU.
// ---------------------------------------------------------------------------
#define LDX_ 260   // 260%64==4 -> conflict-free
#define LDH_ 40    // 2*40%64==16 -> half-wave rows on disjoint banks
__global__ __launch_bounds__(256) void gat_kernel(float* __restrict__ x_io,
                                                  const int* __restrict__ adj,
                                                  const float* __restrict__ Wg,
                                                  const float* __restrict__ a_src,
                                                  const float* __restrict__ a_dst) {
  extern __shared__ float smem[];
  unsigned long long* rowmask = (unsigned long long*)smem;  // 64 * u64
  float* xs   = smem + 128;                                 // 64  * LDX_
  float* hs   = xs + 64 * LDX_;                             // 8*64* LDH_
  float* es   = hs + HEADS_ * 64 * LDH_;                    // 512
  float* ed   = es + 512;                                   // 512
  float* rmax = ed + 512;                                   // 512
  float* rinv = rmax + 512;                                 // 512

  const int g    = blockIdx.x;
  const int tid  = threadIdx.x;
  const int lane = tid & 31;
  const int head = tid >> 5;
  float* xg      = x_io + (size_t)g * N_ * F_;
  const int* ag  = adj + (size_t)g * N_ * N_;

  // x tile via TDM (wave 0); adjacency bitmasks built in parallel
  if (tid < 32) {
    tdm_load_2d_pad4(xg, xs, F_, N_, F_);
  }
  if (tid < N_) {
    unsigned long long m = 0;
    for (int j = 0; j < N_; ++j)
      if (ag[tid * N_ + j] > 0 || j == tid) m |= (1ull << j);
    rowmask[tid] = m;
  }
  if (tid < 32) __builtin_amdgcn_s_wait_tensorcnt(0);
  __syncthreads();

  const int mA = lane & 15;
  const int kk = (lane >> 4) * 2;
  const int nB = lane & 15;

  for (int l = 0; l < LAYERS_; ++l) {
    const float* Wl = Wg + ((size_t)l * HEADS_ + head) * F_ * DH_;

    // ---- phase 1: h[head] = x * W[l,head]   (64x256 * 256x32) ----
    for (int r = 0; r < 4; ++r) {
      const int m0 = r * 16;
      v8f acc0 = {}, acc1 = {};
      const float* arow = xs + (m0 + mA) * LDX_ + kk;
      for (int k = 0; k < F_; k += 4) {
        v2f a, b0, b1;
        a.x = arow[k];
        a.y = arow[k + 1];
        const float* wr0 = Wl + (size_t)(k + kk) * DH_;
        const float* wr1 = wr0 + DH_;
        b0.x = wr0[nB];      b0.y = wr1[nB];
        b1.x = wr0[16 + nB]; b1.y = wr1[16 + nB];
        acc0 = wmma4(a, b0, acc0);
        acc1 = wmma4(a, b1, acc1);
      }
      float* hh = hs + head * 64 * LDH_;
#pragma unroll
      for (int rr = 0; rr < 8; ++rr) {
        const int row = m0 + rr + (lane >> 4) * 8;
        hh[row * LDH_ + nB]      = acc0[rr];
        hh[row * LDH_ + 16 + nB] = acc1[rr];
      }
    }
    __syncthreads();

    // ---- phase 2: es/ed ----
    for (int p = tid; p < HEADS_ * N_; p += 256) {
      const int h = p >> 6, n = p & 63;
      const float* hv = hs + (h * 64 + n) * LDH_;
      const float* as = a_src + ((size_t)l * HEADS_ + h) * DH_;
      const float* ad = a_dst + ((size_t)l * HEADS_ + h) * DH_;
      float s0 = 0.f, s1 = 0.f;
#pragma unroll
      for (int d = 0; d < DH_; ++d) { s0 += hv[d] * as[d]; s1 += hv[d] * ad[d]; }
      es[p] = s0;
      ed[p] = s1;
    }
    __syncthreads();

    // ---- phase 3: masked-softmax row stats ----
    for (int p = tid; p < HEADS_ * N_; p += 256) {
      const int h = p >> 6, i = p & 63;
      const unsigned long long m = rowmask[i];
      const float ei = es[h * 64 + i];
      float mx = -3.0e38f;
      for (int j = 0; j < N_; ++j) {
        float z = ei + ed[h * 64 + j];
        float v = (z > 0.f) ? z : 0.1f * z;          // leaky_relu
        v = ((m >> j) & 1ull) ? v : -1.0e9f;
        mx = fmaxf(mx, v);
      }
      float sum = 0.f;
      for (int j = 0; j < N_; ++j) {
        float z = ei + ed[h * 64 + j];
        float v = (z > 0.f) ? z : 0.1f * z;
        v = ((m >> j) & 1ull) ? v : -1.0e9f;
        sum += __expf(v - mx);
      }
      rmax[p] = mx;
      rinv[p] = 1.f / sum;
    }
    __syncthreads();

    // ---- phase 4: x_new = elu(attn * h), attn A-fragment built on the fly --
    {
      const float* hh = hs + head * 64 * LDH_;
      const float* dh = ed + head * 64;
      for (int r = 0; r < 4; ++r) {
        const int m0 = r * 16;
        const int i = m0 + mA;
        const unsigned long long m = rowmask[i];
        const float ei  = es[head * 64 + i];
        const float mxi = rmax[head * 64 + i];
        const float rvi = rinv[head * 64 + i];
        v8f acc0 = {}, acc1 = {};
        for (int k = 0; k < N_; k += 4) {
          const int j0 = k + kk, j1 = k + kk + 1;
          float z0 = ei + dh[j0]; z0 = (z0 > 0.f) ? z0 : 0.1f * z0;
          float z1 = ei + dh[j1]; z1 = (z1 > 0.f) ? z1 : 0.1f * z1;
          z0 = ((m >> j0) & 1ull) ? z0 : -1.0e9f;
          z1 = ((m >> j1) & 1ull) ? z1 : -1.0e9f;
          v2f a, b0, b1;
          a.x = __expf(z0 - mxi) * rvi;
          a.y = __expf(z1 - mxi) * rvi;
          b0.x = hh[j0 * LDH_ + nB];      b0.y = hh[j1 * LDH_ + nB];
          b1.x = hh[j0 * LDH_ + 16 + nB]; b1.y = hh[j1 * LDH_ + 16 + nB];
          acc0 = wmma4(a, b0, acc0);
          acc1 = wmma4(a, b1, acc1);
        }
#pragma unroll
        for (int rr = 0; rr < 8; ++rr) {
          const int row = m0 + rr + (lane >> 4) * 8;
          float v0 = acc0[rr]; v0 = (v0 > 0.f) ? v0 : expm1f(v0);  // ELU
          float v1 = acc1[rr]; v1 = (v1 > 0.f) ? v1 : expm1f(v1);
          xs[row * LDX_ + head * DH_ + nB]      = v0;
          xs[row * LDX_ + head * DH_ + 16 + nB] = v1;
        }
      }
    }
    __syncthreads();
  }

  // write back (in-place; TDM store has no de-padding, so store manually)
  for (int i = tid; i < N_ * F_; i += 256)
    xg[i] = xs[(i >> 8) * LDX_ + (i & 255)];
}

// ---------------------------------------------------------------------------
// Kernel 3: father_feat[b,e,f] = mean_n child_out[b,e,n,f]
// ---------------------------------------------------------------------------
__global__ __launch_bounds__(256) void mean_kernel(const float* __restrict__ x,
                                                   float* __restrict__ out) {
  const int be = blockIdx.x;
  const int f  = threadIdx.x;
  const float* p = x + (size_t)be * N_ * F_ + f;
  float s = 0.f;
#pragma unroll 8
  for (int n = 0; n < N_; ++n) s += p[n * F_];
  out[(size_t)be * F_ + f] = s * (1.f / (float)N_);
}

// ---------------------------------------------------------------------------
// Kernel 4: BiLSTM. grid(B,2): one block per (batch, direction).
// 1024 threads, one gate row each; h/c/gates in LDS; 64 sequential steps.
// ---------------------------------------------------------------------------
__global__ __launch_bounds__(1024) void lstm_kernel(const float* __restrict__ seq_in, // [B][E][F]
                                                    const float* __restrict__ w_ih,   // [2][4H][F]
                                                    const float* __restrict__ w_hh,   // [2][4H][H]
                                                    const float* __restrict__ bg,     // [2][4H]
                                                    float* __restrict__ seq_out) {    // [B][E][2H]
  __shared__ float h[HID_], c[HID_], xt[F_], gates[4 * HID_];
  const int b   = blockIdx.x;
  const int dir = blockIdx.y;
  const int tid = threadIdx.x;
  if (tid < HID_) { h[tid] = 0.f; c[tid] = 0.f; }
  const float* wi = w_ih + ((size_t)dir * 4 * HID_ + tid) * F_;
  const float* wh = w_hh + ((size_t)dir * 4 * HID_ + tid) * HID_;
  const float bb  = bg[dir * 4 * HID_ + tid];
  __syncthreads();
  for (int s = 0; s < E_; ++s) {
    const int t = dir ? (E_ - 1 - s) : s;
    if (tid < F_) xt[tid] = seq_in[((size_t)b * E_ + t) * F_ + tid];
    __syncthreads();
    float acc = bb;
    for (int k = 0; k < F_; ++k)   acc += wi[k] * xt[k];
    for (int k = 0; k < HID_; ++k) acc += wh[k] * h[k];
    gates[tid] = acc;
    __syncthreads();
    if (tid < HID_) {
      const float ig = 1.f / (1.f + __expf(-gates[tid]));
      const float fg = 1.f / (1.f + __expf(-gates[HID_ + tid]));
      const float gg = tanhf(gates[2 * HID_ + tid]);
      const float og = 1.f / (1.f + __expf(-gates[3 * HID_ + tid]));
      const float cn = fg * c[tid] + ig * gg;
      c[tid] = cn;
      const float hn = og * tanhf(cn);
      h[tid] = hn;
      seq_out[((size_t)b * E_ + t) * (2 * HID_) + dir * HID_ + tid] = hn;
    }
    __syncthreads();
  }
}

// ---------------------------------------------------------------------------
// Kernel 5: self-attention head + classifier. One block per batch.
// feat = mean_q(softmax(seq seq^T / sqrt(d)) seq) = (colmean attn)^T seq
// ---------------------------------------------------------------------------
#define LDQ_ 514   // 512 padded; 514%64==2 -> conflict-free column walks
__global__ __launch_bounds__(256) void head_kernel(const float* __restrict__ seq,   // [B][E][2H]
                                                   const float* __restrict__ cls_w, // [2H][2]
                                                   const float* __restrict__ cls_b, // [2]
                                                   float* __restrict__ out) {       // pred[B][2] ++ feat[B][2H]
  extern __shared__ float smem[];
  float* s    = smem;                 // E_ * LDQ_
  float* sc   = s + E_ * LDQ_;        // E_ * E_
  float* cw   = sc + E_ * E_;         // E_
  float* feat = cw + E_;              // 2H
  const int b   = blockIdx.x;
  const int tid = threadIdx.x;
  const float* sq = seq + (size_t)b * E_ * (2 * HID_);

  for (int i = tid; i < E_ * 2 * HID_; i += 256)
    s[(i >> 9) * LDQ_ + (i & 511)] = sq[i];
  __syncthreads();

  const float scale = 1.f / sqrtf((float)(2 * HID_));
  for (int p = tid; p < E_ * E_; p += 256) {
    const int q = p >> 6, k = p & 63;
    const float* a  = s + q * LDQ_;
    const float* bb = s + k * LDQ_;
    float acc = 0.f;
    for (int d = 0; d < 2 * HID_; ++d) acc += a[d] * bb[d];
    sc[p] = acc * scale;
  }
  __syncthreads();

  if (tid < E_) {  // softmax per row
    float* row = sc + tid * E_;
    float mx = row[0];
    for (int k = 1; k < E_; ++k) mx = fmaxf(mx, row[k]);
    float sum = 0.f;
    for (int k = 0; k < E_; ++k) { const float e = __expf(row[k] - mx); row[k] = e; sum += e; }
    const float inv = 1.f / sum;
    for (int k = 0; k < E_; ++k) row[k] *= inv;
  }
  __syncthreads();

  if (tid < E_) {  // column means of attn
    float acc = 0.f;
    for (int q = 0; q < E_; ++q) acc += sc[q * E_ + tid];
    cw[tid] = acc * (1.f / (float)E_);
  }
  __syncthreads();

  for (int d = tid; d < 2 * HID_; d += 256) {
    float acc = 0.f;
    for (int k = 0; k < E_; ++k) acc += cw[k] * s[k * LDQ_ + d];
    feat[d] = acc;
    out[2 * B_ + b * (2 * HID_) + d] = acc;   // feat section of d_out
  }
  __syncthreads();

  if (tid < 2) {
    float acc = cls_b[tid];
    for (int d = 0; d < 2 * HID_; ++d) acc += feat[d] * cls_w[d * 2 + tid];
    out[b * 2 + tid] = acc;                   // pred section of d_out
  }
}

// ---------------------------------------------------------------------------
extern "C" void kernel_launch(void* const* d_in, const int* in_sizes, int n_in,
                              void* d_out, int out_size, void* d_ws, size_t ws_size,
                              hipStream_t stream) {
  (void)in_sizes; (void)n_in; (void)out_size; (void)ws_size;
  const float* child_feat = (const float*)d_in[0];
  const int*   child_adj  = (const int*)d_in[1];
  const int*   father_adj = (const int*)d_in[2];
  const float* fc_w       = (const float*)d_in[3];
  const float* fc_b       = (const float*)d_in[4];
  const float* child_W    = (const float*)d_in[5];
  const float* child_as   = (const float*)d_in[6];
  const float* child_ad   = (const float*)d_in[7];
  const float* father_W   = (const float*)d_in[8];
  const float* father_as  = (const float*)d_in[9];
  const float* father_ad  = (const float*)d_in[10];
  const float* w_ih       = (const float*)d_in[11];
  const float* w_hh       = (const float*)d_in[12];
  const float* lstm_b     = (const float*)d_in[13];
  const float* cls_w      = (const float*)d_in[14];
  const float* cls_b      = (const float*)d_in[15];
  float* out = (float*)d_out;

  float* xbuf  = (float*)d_ws;                          // [B*E*N][F]  33.6 MB
  float* ffeat = xbuf + (size_t)B_ * E_ * N_ * F_;      // [B][E][F]    0.5 MB
  float* seq   = ffeat + (size_t)B_ * E_ * F_;          // [B][E][2H]   1.0 MB

  // 1. FC GEMM (WMMA f32, TDM-staged A tiles)
  fc_kernel<<<dim3(2048, 2), 256, 0, stream>>>(child_feat, fc_w, fc_b, xbuf);

  const size_t gat_lds = 512 + (size_t)64 * LDX_ * 4 +
                         (size_t)HEADS_ * 64 * LDH_ * 4 + 4 * 512 * 4; // ~157 KB
  // 2. Child GATs, in place on xbuf (one WG per (b,e) graph)
  gat_kernel<<<dim3(B_ * E_), 256, gat_lds, stream>>>(xbuf, child_adj, child_W,
                                                      child_as, child_ad);
  // 3. Mean over N
  mean_kernel<<<dim3(B_ * E_), 256, 0, stream>>>(xbuf, ffeat);
  // 4. Father GAT, in place on ffeat (one WG per batch graph)
  gat_kernel<<<dim3(B_), 256, gat_lds, stream>>>(ffeat, father_adj, father_W,
                                                 father_as, father_ad);
  // 5. BiLSTM
  lstm_kernel<<<dim3(B_, 2), 1024, 0, stream>>>(ffeat, w_ih, w_hh, lstm_b, seq);
  // 6. Attention head + classifier
  const size_t head_lds = (size_t)(E_ * LDQ_ + E_ * E_ + E_ + 2 * HID_) * 4; // ~150 KB
  head_kernel<<<dim3(B_), 256, head_lds, stream>>>(seq, cls_w, cls_b, out);
}